// LSTM_11089605558768
// MI455X (gfx1250) — compile-verified
//
#include <hip/hip_runtime.h>
#include <hip/hip_bf16.h>
#include <math.h>

// Problem constants (from reference): T=16384, B=1, IN=32, H=512, OUT=1
#define T_STEPS 16384
#define IN_DIM  32
#define H_DIM   512
#define G4      (4 * H_DIM)   // 2048 gate rows (i,f,g,o)

// Recurrence sharding: 32 persistent workgroups, each owns 16 hidden units
// -> 64 rows of W_hh (4 gates x 16 units) = 64*512 fp32 = 128 KB in LDS.
#define NWG   32
#define TPB2  512
#define UPW   16              // hidden units per WG
#define ROWS  64              // W_hh rows per WG
#define PPR   8               // threads per row (512 / 64)
#define KCH   64              // K elements per thread (512 / 8)
#define CHS   66              // padded chunk stride (words) -> conflict-free banks
#define RSTR  (PPR * CHS)     // padded row stride = 528 words

typedef float v2f __attribute__((ext_vector_type(2)));
typedef float v8f __attribute__((ext_vector_type(8)));

// ---------------------------------------------------------------------------
// Kernel 1: x_gates[T,2048] = X[T,32] @ W_ih^T + (b_ih + b_hh), fp32 WMMA.
// One 16x16 output tile per wave; K=32 as 8 x V_WMMA_F32_16X16X4_F32.
// Also resets the grid-barrier counter and zeroes the h double-buffer
// (must be re-done every launch: harness does not re-poison between replays).
// ---------------------------------------------------------------------------
__global__ __launch_bounds__(256) void xgates_gemm(
    const float* __restrict__ X, const float* __restrict__ W_ih,
    const float* __restrict__ b_ih, const float* __restrict__ b_hh,
    float* __restrict__ xg, float* __restrict__ h_buf,
    unsigned* __restrict__ counter)
{
  const int wave  = threadIdx.x >> 5;
  const int lane  = threadIdx.x & 31;
  const int gtile = blockIdx.x * 8 + wave;       // (T/16)*(2048/16) = 131072 tiles
  const int mt = gtile >> 7;                     // / 128 n-tiles
  const int nt = gtile & 127;
  const int m0 = mt * 16, n0 = nt * 16;

  const int half = lane >> 4;                    // ISA: lanes 16-31 hold K+2/K+3
  const int mrow = lane & 15;
  const int ncol = n0 + (lane & 15);

  v8f c = {0.f, 0.f, 0.f, 0.f, 0.f, 0.f, 0.f, 0.f};
#pragma unroll
  for (int k0 = 0; k0 < IN_DIM; k0 += 4) {
    const int ka = k0 + 2 * half;
    v2f a, b;
    // A 16x4 fp32 layout: v0 = A[m][ka], v1 = A[m][ka+1]
    a.x = X[(size_t)(m0 + mrow) * IN_DIM + ka];
    a.y = X[(size_t)(m0 + mrow) * IN_DIM + ka + 1];
    // B 4x16 fp32 (mirror of A): B[k][n] = W_ih[n][k]
    b.x = W_ih[(size_t)ncol * IN_DIM + ka];
    b.y = W_ih[(size_t)ncol * IN_DIM + ka + 1];
    c = __builtin_amdgcn_wmma_f32_16x16x4_f32(
        /*neg_a=*/false, a, /*neg_b=*/false, b,
        /*c_mod=*/(short)0, c, /*reuse_a=*/false, /*reuse_b=*/false);
  }

  const float bias = b_ih[ncol] + b_hh[ncol];
#pragma unroll
  for (int i = 0; i < 8; ++i) {                  // C/D: VGPR i holds M = i + 8*half
    const int m = m0 + i + 8 * half;
    xg[(size_t)m * G4 + ncol] = c[i] + bias;
  }

  // Per-launch state init (kernel 2 runs after us in stream order).
  if (blockIdx.x == 0) {
    if (threadIdx.x == 0) *counter = 0u;
    for (int i = threadIdx.x; i < 2 * H_DIM; i += 256) h_buf[i] = 0.f;
  }
}

// ---------------------------------------------------------------------------
// Device-scope generation barrier across NWG persistent workgroups.
// ---------------------------------------------------------------------------
__device__ __forceinline__ void grid_sync(unsigned* counter)
{
  __syncthreads();
  if (threadIdx.x == 0) {
    __threadfence();  // make this wave's h stores device-visible
    unsigned prev = __hip_atomic_fetch_add(counter, 1u, __ATOMIC_ACQ_REL,
                                           __HIP_MEMORY_SCOPE_AGENT);
    unsigned target = prev - (prev % NWG) + NWG;
    while (__hip_atomic_load(counter, __ATOMIC_ACQUIRE,
                             __HIP_MEMORY_SCOPE_AGENT) < target) {
      __builtin_amdgcn_s_sleep(2);
    }
  }
  __syncthreads();
}

// ---------------------------------------------------------------------------
// Kernel 2: persistent LSTM recurrence. 32 WGs x 512 threads; each WG keeps
// its 64x512 fp32 W_hh shard in LDS (bank-conflict-free padded layout) for
// all 16384 steps; h (512 fp32) is exchanged through L2 each step with a
// device barrier. Gate order i,f,g,o as in PyTorch. Final step: WG0 reduces
// dot(W_lin, h_T) + b_lin into d_out[0].
// ---------------------------------------------------------------------------
__global__ __launch_bounds__(TPB2) void lstm_recurrence(
    const float* __restrict__ W_hh, const float* __restrict__ xg,
    const float* __restrict__ W_lin, const float* __restrict__ b_lin,
    float* __restrict__ out, float* __restrict__ h_buf,
    unsigned* __restrict__ counter)
{
  __shared__ float lds_w[ROWS * RSTR];   // 64 rows x 528 words = 132 KB
  __shared__ float lds_h[PPR * CHS];     // padded h copy
  __shared__ float gbuf[ROWS];           // per-row gate preactivations
  __shared__ float cbuf[UPW];            // cell state for this WG's units
  __shared__ float red[TPB2];            // final reduction

  const int tid = threadIdx.x;
  const int wg  = blockIdx.x;
  const int u0  = wg * UPW;              // first hidden unit owned by this WG

  // Preload W_hh shard into padded LDS layout:
  // element (r, k) -> lds_w[r*RSTR + (k/64)*CHS + (k%64)]
  for (int idx = tid; idx < ROWS * H_DIM; idx += TPB2) {
    const int r = idx >> 9;              // local row 0..63
    const int k = idx & 511;
    const int g = r >> 4;                // gate 0..3
    const int u = r & 15;                // unit 0..15
    lds_w[r * RSTR + (k >> 6) * CHS + (k & 63)] =
        W_hh[(size_t)(g * H_DIM + u0 + u) * H_DIM + k];
  }
  if (tid < UPW) cbuf[tid] = 0.f;
  __syncthreads();

  const int r = tid >> 3;                // my row (0..63)
  const int p = tid & 7;                 // my K-chunk (0..7)
  const float* wrow = lds_w + r * RSTR + p * CHS;

  for (int t = 0; t < T_STEPS; ++t) {
    // Fetch current h (written by all WGs last step) into LDS, padded layout.
    {
      const float* hsrc = h_buf + (size_t)(t & 1) * H_DIM;
      float hv = __hip_atomic_load(&hsrc[tid], __ATOMIC_RELAXED,
                                   __HIP_MEMORY_SCOPE_AGENT);
      lds_h[(tid >> 6) * CHS + (tid & 63)] = hv;
    }
    // Prefetch next step's x_gate slice (4 x 64B cache lines for this WG).
    if (t + 1 < T_STEPS && tid < 4)
      __builtin_prefetch(xg + (size_t)(t + 1) * G4 + tid * H_DIM + u0, 0, 1);
    __syncthreads();

    // Partial dot: 64 fp32 MACs per thread, all 32 lanes on distinct LDS banks.
    float acc = 0.f;
    const float* hh = lds_h + p * CHS;
#pragma unroll
    for (int k = 0; k < KCH; ++k) acc = fmaf(wrow[k], hh[k], acc);
    // Reduce the 8 chunk-partials (consecutive lanes) inside the wave.
    acc += __shfl_xor(acc, 1, 32);
    acc += __shfl_xor(acc, 2, 32);
    acc += __shfl_xor(acc, 4, 32);
    if (p == 0) {
      const int g = r >> 4, u = r & 15;
      gbuf[r] = acc + xg[(size_t)t * G4 + g * H_DIM + u0 + u];
    }
    __syncthreads();

    // Gate nonlinearities + state update for this WG's 16 units.
    if (tid < UPW) {
      const int u = tid;
      const float gi = gbuf[u];
      const float gf = gbuf[UPW + u];
      const float gg = gbuf[2 * UPW + u];
      const float go = gbuf[3 * UPW + u];
      const float i_ = 1.f / (1.f + expf(-gi));
      const float f_ = 1.f / (1.f + expf(-gf));
      const float o_ = 1.f / (1.f + expf(-go));
      const float cn = f_ * cbuf[u] + i_ * tanhf(gg);
      cbuf[u] = cn;
      const float hn = o_ * tanhf(cn);
      __hip_atomic_store(&h_buf[(size_t)((t + 1) & 1) * H_DIM + u0 + u], hn,
                         __ATOMIC_RELAXED, __HIP_MEMORY_SCOPE_AGENT);
    }
    grid_sync(counter);                  // release h, acquire everyone else's
  }

  // predictions[-1] = dot(W_lin, h_T) + b_lin ; h_T lives in parity buffer 0.
  if (wg == 0) {
    float v = W_lin[tid] * __hip_atomic_load(&h_buf[tid], __ATOMIC_RELAXED,
                                             __HIP_MEMORY_SCOPE_AGENT);
    red[tid] = v;
    __syncthreads();
    for (int s = TPB2 / 2; s > 0; s >>= 1) {
      if (tid < s) red[tid] += red[tid + s];
      __syncthreads();
    }
    if (tid == 0) out[0] = red[0] + b_lin[0];
  }
}

// ---------------------------------------------------------------------------
// Host launch. Workspace layout: x_gates (128 MB) | h double-buffer | counter.
// ---------------------------------------------------------------------------
extern "C" void kernel_launch(void* const* d_in, const int* in_sizes, int n_in,
                              void* d_out, int out_size, void* d_ws, size_t ws_size,
                              hipStream_t stream)
{
  const float* X     = (const float*)d_in[0];   // [T,1,IN]
  const float* W_ih  = (const float*)d_in[1];   // [2048,32]
  const float* W_hh  = (const float*)d_in[2];   // [2048,512]
  const float* b_ih  = (const float*)d_in[3];   // [2048]
  const float* b_hh  = (const float*)d_in[4];   // [2048]
  const float* W_lin = (const float*)d_in[5];   // [1,512]
  const float* b_lin = (const float*)d_in[6];   // [1]
  float* out = (float*)d_out;                   // [1]

  float*    xg      = (float*)d_ws;                       // T*2048 fp32
  float*    h_buf   = xg + (size_t)T_STEPS * G4;          // 2*512 fp32
  unsigned* counter = (unsigned*)(h_buf + 2 * H_DIM);     // 1 u32

  // Kernel 1: (T/16)*(2048/16) tiles, 8 waves (tiles) per 256-thread block.
  const int tiles  = (T_STEPS / 16) * (G4 / 16);
  xgates_gemm<<<tiles / 8, 256, 0, stream>>>(X, W_ih, b_ih, b_hh,
                                             xg, h_buf, counter);
  // Kernel 2: 32 persistent WGs (one per WGP; MI455X has far more than 32
  // WGPs, so all are co-resident and the device barrier cannot deadlock).
  lstm_recurrence<<<NWG, TPB2, 0, stream>>>(W_hh, xg, W_lin, b_lin,
                                            out, h_buf, counter);
}